// GraphEncoder_60911226192365
// MI455X (gfx1250) — compile-verified
//
#include <hip/hip_runtime.h>

typedef __attribute__((ext_vector_type(2))) float v2f;
typedef __attribute__((ext_vector_type(8))) float v8f;

#define N_NODES 50000
#define IN_CH   128
#define HID_CH  256
#define N_EDGES 800000

// ---------------- Phase 1: zero the accumulators in workspace ----------------
__global__ void sage_zero_ws(float* __restrict__ ws, int n) {
    int i = blockIdx.x * blockDim.x + threadIdx.x;
    if (i < n) ws[i] = 0.0f;
}

// ---------------- Phase 2: scatter-sum of x[src] into agg[dst], count deg ----
// One wave (32 lanes) per edge; each lane handles 4 contiguous channels as a
// float4 load + 4 global f32 atomic adds (L2-resident: x and agg fit in 192MB L2).
__global__ void sage_scatter(const float* __restrict__ x,
                             const long long* __restrict__ ei,
                             float* __restrict__ agg,
                             float* __restrict__ deg) {
    int t    = blockIdx.x * blockDim.x + threadIdx.x;
    int e    = t >> 5;
    int lane = t & 31;
    if (e >= N_EDGES) return;
    long long src = ei[e];
    long long dst = ei[N_EDGES + e];
    const float4 v = *((const float4*)(x + src * IN_CH) + lane);
    float* a = agg + dst * IN_CH + lane * 4;
    atomicAdd(a + 0, v.x);
    atomicAdd(a + 1, v.y);
    atomicAdd(a + 2, v.z);
    atomicAdd(a + 3, v.w);
    if (lane == 0) atomicAdd(deg + dst, 1.0f);
}

// ---------------- Phase 3: fused GEMM via full-fp32 WMMA ---------------------
// out = relu( (agg_sum/deg) @ W_l + b_l + x @ W_r )
// One wave per 16x16 output tile; K looped in steps of 4 with
// v_wmma_f32_16x16x4_f32 (exact f32, no precision loss).
//
// Layouts (CDNA5 ISA 7.12.2):
//   A 16x4 f32:  lane L -> row M = L&15; VGPR0 = K+(hi?2:0), VGPR1 = K+1+(hi?2:0)
//   B 4x16 f32:  lane L -> col N = L&15; VGPR0 = K+(hi?2:0), VGPR1 = K+1+(hi?2:0)
//   C/D 16x16:   VGPR r -> row M = r + (hi?8:0), col N = L&15
__global__ void sage_gemm_wmma(const float* __restrict__ x,
                               const float* __restrict__ Wl,
                               const float* __restrict__ bl,
                               const float* __restrict__ Wr,
                               const float* __restrict__ agg,
                               const float* __restrict__ deg,
                               float* __restrict__ out) {
    const int waves_per_block = blockDim.x >> 5;
    int wave = blockIdx.x * waves_per_block + (threadIdx.x >> 5);
    int lane = threadIdx.x & 31;

    const int nTilesN = HID_CH / 16;          // 16
    int mTile = wave / nTilesN;
    int nTile = wave - mTile * nTilesN;
    if (mTile >= N_NODES / 16) return;        // wave-uniform (never splits EXEC)

    int m0  = mTile * 16;
    int n0  = nTile * 16;
    int lo  = lane & 15;
    int koff = (lane >> 4) ? 2 : 0;           // high half-wave handles K+2,K+3

    int arow = m0 + lo;                       // A row owned by this lane
    int col  = n0 + lo;                       // B/C/D column owned by this lane

    float invd = 1.0f / fmaxf(deg[arow], 1.0f);
    float bias = bl[col];

    v8f acc;
#pragma unroll
    for (int r = 0; r < 8; ++r) acc[r] = bias;

    const float* aggRow = agg + (long)arow * IN_CH;
    const float* xRow   = x   + (long)arow * IN_CH;

    // K = 0..127 : (agg_sum * invd) @ W_l
    for (int k = 0; k < IN_CH; k += 4) {
        int ka = k + koff;
        v2f a, b;
        a.x = aggRow[ka]     * invd;
        a.y = aggRow[ka + 1] * invd;
        b.x = Wl[(long)ka       * HID_CH + col];
        b.y = Wl[(long)(ka + 1) * HID_CH + col];
        acc = __builtin_amdgcn_wmma_f32_16x16x4_f32(
            false, a, false, b, (short)0, acc, false, false);
    }
    // K = 128..255 : x @ W_r
    for (int k = 0; k < IN_CH; k += 4) {
        int ka = k + koff;
        v2f a, b;
        a.x = xRow[ka];
        a.y = xRow[ka + 1];
        b.x = Wr[(long)ka       * HID_CH + col];
        b.y = Wr[(long)(ka + 1) * HID_CH + col];
        acc = __builtin_amdgcn_wmma_f32_16x16x4_f32(
            false, a, false, b, (short)0, acc, false, false);
    }

#pragma unroll
    for (int r = 0; r < 8; ++r) {
        int orow = m0 + r + ((lane >> 4) ? 8 : 0);
        out[(long)orow * HID_CH + col] = fmaxf(acc[r], 0.0f);
    }
}

// ---------------- Host launcher ----------------------------------------------
extern "C" void kernel_launch(void* const* d_in, const int* in_sizes, int n_in,
                              void* d_out, int out_size, void* d_ws, size_t ws_size,
                              hipStream_t stream) {
    const float*     x   = (const float*)d_in[0];
    const long long* ei  = (const long long*)d_in[1];   // int64 [2, E]
    const float*     Wl  = (const float*)d_in[2];
    const float*     bl  = (const float*)d_in[3];
    const float*     Wr  = (const float*)d_in[4];
    float*           out = (float*)d_out;

    float* agg = (float*)d_ws;                          // [N_NODES, IN_CH]
    float* deg = agg + (size_t)N_NODES * IN_CH;         // [N_NODES]

    // Phase 1: zero agg + deg
    {
        int n = N_NODES * IN_CH + N_NODES;
        int blocks = (n + 255) / 256;
        sage_zero_ws<<<blocks, 256, 0, stream>>>((float*)d_ws, n);
    }
    // Phase 2: scatter
    {
        long long threads = (long long)N_EDGES * 32;
        int blocks = (int)((threads + 255) / 256);
        sage_scatter<<<blocks, 256, 0, stream>>>(x, ei, agg, deg);
    }
    // Phase 3: fused WMMA GEMM  (50000 tiles = 6250 blocks * 8 waves)
    {
        int totalWaves = (N_NODES / 16) * (HID_CH / 16);
        int blocks = (totalWaves + 7) / 8;
        sage_gemm_wmma<<<blocks, 256, 0, stream>>>(x, Wl, bl, Wr, agg, deg, out);
    }
}